// CorrAttention_46712064311835
// MI455X (gfx1250) — compile-verified
//
#include <hip/hip_runtime.h>
#include <math.h>

// ---------------------------------------------------------------------------
// CorrAttention for MI455X (gfx1250, wave32, WMMA)
// B=16, N=512, n=32, M=B*N=8192, F=n*n=1024
// ---------------------------------------------------------------------------

typedef __attribute__((ext_vector_type(2))) float v2f;
typedef __attribute__((ext_vector_type(8))) float v8f;

#define WMMA_F32_16x16x4(a, b, c) \
  __builtin_amdgcn_wmma_f32_16x16x4_f32(false, (a), false, (b), (short)0, (c), false, false)

#define NMAT 8192   // B*N
#define NB   16
#define NN   512
#define NF   1024

// ---------------------------------------------------------------------------
// Kernel 1: C_w = (I - X) (I + X)^{-1},  X = W - W^T   (3 weights)
// One 256-thread block per weight; Gauss-Jordan in LDS.
// ---------------------------------------------------------------------------
__global__ __launch_bounds__(256) void cayley_kernel(const float* __restrict__ qw,
                                                     const float* __restrict__ kw,
                                                     const float* __restrict__ vw,
                                                     float* __restrict__ Cout) {
  __shared__ float X[32][32];
  __shared__ float Mm[32][33];
  __shared__ float Inv[32][33];
  __shared__ float fac[32];
  __shared__ float pivinv;
  const float* W = (blockIdx.x == 0) ? qw : ((blockIdx.x == 1) ? kw : vw);
  const int tid = threadIdx.x;

  for (int e = tid; e < 1024; e += 256) {
    int i = e >> 5, j = e & 31;
    float xv = W[i * 32 + j] - W[j * 32 + i];
    X[i][j]  = xv;
    Mm[i][j] = xv + ((i == j) ? 1.f : 0.f);
    Inv[i][j] = (i == j) ? 1.f : 0.f;
  }
  __syncthreads();

  for (int k = 0; k < 32; ++k) {
    if (tid == 0) pivinv = 1.f / Mm[k][k];
    __syncthreads();
    if (tid < 32)      Mm[k][tid]       *= pivinv;
    else if (tid < 64) Inv[k][tid - 32] *= pivinv;
    __syncthreads();
    if (tid < 32) fac[tid] = (tid == k) ? 0.f : Mm[tid][k];
    __syncthreads();
    for (int e = tid; e < 2048; e += 256) {
      int i = e >> 6, j = e & 63;
      if (i == k) continue;
      float f = fac[i];
      if (j < 32) Mm[i][j]        -= f * Mm[k][j];
      else        Inv[i][j - 32]  -= f * Inv[k][j - 32];
    }
    __syncthreads();
  }

  // C = (I - X) * Inv
  for (int e = tid; e < 1024; e += 256) {
    int i = e >> 5, j = e & 31;
    float acc = 0.f;
    for (int kk = 0; kk < 32; ++kk) {
      float a = ((i == kk) ? 1.f : 0.f) - X[i][kk];
      acc += a * Inv[kk][j];
    }
    Cout[blockIdx.x * 1024 + i * 32 + j] = acc;
  }
}

// ---------------------------------------------------------------------------
// Kernel 2: xl = olm_diffeo(sym_logm(x)) = V log(w) V^T with zero diagonal.
// One wave32 per 32x32 matrix; parallel cyclic Jacobi (16 disjoint pairs per
// round, 31 rounds per sweep, 8 sweeps).
// ---------------------------------------------------------------------------
__global__ __launch_bounds__(32) void jacobi_logm_kernel(const float* __restrict__ x,
                                                         float* __restrict__ xl) {
  __shared__ float A[32][33];
  __shared__ float V[32][33];
  __shared__ float cc[16], ss[16];
  __shared__ int   pp[16], qq[16];
  __shared__ float lw[32];
  const int lane = threadIdx.x;
  const float* xm = x + (size_t)blockIdx.x * 1024;

  for (int i = 0; i < 32; ++i) {
    A[i][lane] = xm[i * 32 + lane];
    V[i][lane] = (i == lane) ? 1.f : 0.f;
  }
  __syncthreads();

  for (int sweep = 0; sweep < 8; ++sweep) {
    for (int r = 0; r < 31; ++r) {
      if (lane < 16) {
        int p, q;
        if (lane == 0) { p = 31; q = r % 31; }
        else { p = (r + lane) % 31; q = (r - lane + 31) % 31; }
        if (p > q) { int t = p; p = q; q = t; }
        float apq = A[p][q];
        float c = 1.f, s = 0.f;
        if (fabsf(apq) > 1e-12f) {
          float theta = (A[q][q] - A[p][p]) / (2.f * apq);
          float t = 1.f / (fabsf(theta) + sqrtf(theta * theta + 1.f));
          if (theta < 0.f) t = -t;
          c = rsqrtf(t * t + 1.f);
          s = t * c;
        }
        cc[lane] = c; ss[lane] = s; pp[lane] = p; qq[lane] = q;
      }
      __syncthreads();
      // row phase: (J^T A): rows p,q  (16 disjoint pairs, lane = column)
      for (int t = 0; t < 16; ++t) {
        int p = pp[t], q = qq[t];
        float c = cc[t], s = ss[t];
        float ap = A[p][lane], aq = A[q][lane];
        A[p][lane] = c * ap - s * aq;
        A[q][lane] = s * ap + c * aq;
      }
      __syncthreads();
      // column phase: (A J) and (V J): cols p,q (lane = row)
      for (int t = 0; t < 16; ++t) {
        int p = pp[t], q = qq[t];
        float c = cc[t], s = ss[t];
        float ap = A[lane][p], aq = A[lane][q];
        A[lane][p] = c * ap - s * aq;
        A[lane][q] = s * ap + c * aq;
        float vp = V[lane][p], vq = V[lane][q];
        V[lane][p] = c * vp - s * vq;
        V[lane][q] = s * vp + c * vq;
      }
      __syncthreads();
    }
  }

  lw[lane] = logf(fmaxf(A[lane][lane], 1e-20f));
  __syncthreads();

  // s[i][j] = sum_k V[i][k] * lw[k] * V[j][k]; lane owns column j = lane
  float vj[32];
#pragma unroll
  for (int k = 0; k < 32; ++k) vj[k] = V[lane][k] * lw[k];

  float* outm = xl + (size_t)blockIdx.x * 1024;
  for (int i = 0; i < 32; ++i) {
    float acc = 0.f;
#pragma unroll
    for (int k = 0; k < 32; ++k) acc += V[i][k] * vj[k];
    outm[i * 32 + lane] = (i == lane) ? 0.f : acc;  // tril(-1) + mirror of symmetric s
  }
}

// ---------------------------------------------------------------------------
// Kernel 3: trans_base: y2 = C y C^T, then lower-triangle mirror w/ zero diag.
// One wave per (matrix, weight). 32x32 GEMMs as 2x2 tiles of wmma f32 16x16x4.
// ---------------------------------------------------------------------------
__global__ __launch_bounds__(32) void transbase_kernel(const float* __restrict__ xl,
                                                       const float* __restrict__ C,
                                                       float* __restrict__ qkv) {
  __shared__ float Y[32][33];
  __shared__ float CL[32][33];
  __shared__ float T[32][33];
  __shared__ float T2[32][33];
  const int lane = threadIdx.x;
  const int nIdx = lane & 15;
  const int hi   = lane >> 4;
  const int m = blockIdx.x, w = blockIdx.y;
  const float* ym = xl + (size_t)m * 1024;
  const float* Cw = C + (size_t)w * 1024;

  for (int i = 0; i < 32; ++i) {
    Y[i][lane]  = ym[i * 32 + lane];
    CL[i][lane] = Cw[i * 32 + lane];
  }
  __syncthreads();

  // T = C * Y
  for (int ti = 0; ti < 2; ++ti)
    for (int tj = 0; tj < 2; ++tj) {
      v8f acc = {};
#pragma unroll
      for (int kk = 0; kk < 8; ++kk) {
        int k0 = kk * 4 + hi * 2;
        v2f a; a.x = CL[ti * 16 + nIdx][k0]; a.y = CL[ti * 16 + nIdx][k0 + 1];
        v2f b; b.x = Y[k0][tj * 16 + nIdx];  b.y = Y[k0 + 1][tj * 16 + nIdx];
        acc = WMMA_F32_16x16x4(a, b, acc);
      }
#pragma unroll
      for (int r = 0; r < 8; ++r)
        T[ti * 16 + r + hi * 8][tj * 16 + nIdx] = acc[r];
    }
  __syncthreads();

  // T2 = T * C^T   (B[k][n] = C[n][k])
  for (int ti = 0; ti < 2; ++ti)
    for (int tj = 0; tj < 2; ++tj) {
      v8f acc = {};
#pragma unroll
      for (int kk = 0; kk < 8; ++kk) {
        int k0 = kk * 4 + hi * 2;
        v2f a; a.x = T[ti * 16 + nIdx][k0];       a.y = T[ti * 16 + nIdx][k0 + 1];
        v2f b; b.x = CL[tj * 16 + nIdx][k0];      b.y = CL[tj * 16 + nIdx][k0 + 1];
        acc = WMMA_F32_16x16x4(a, b, acc);
      }
#pragma unroll
      for (int r = 0; r < 8; ++r)
        T2[ti * 16 + r + hi * 8][tj * 16 + nIdx] = acc[r];
    }
  __syncthreads();

  // tril(-1) + mirror
  float* outm = qkv + ((size_t)w * NMAT + m) * 1024;
  for (int i = 0; i < 32; ++i) {
    float lo = T2[i][lane];
    float up = T2[lane][i];
    float val = (i > lane) ? lo : ((i < lane) ? up : 0.f);
    outm[i * 32 + lane] = val;
  }
}

// ---------------------------------------------------------------------------
// Kernel 4: squared Frobenius norms of q rows (nrm[0..8191]) and k rows
// (nrm[8192..16383]).
// ---------------------------------------------------------------------------
__global__ __launch_bounds__(256) void norms_kernel(const float* __restrict__ qkv,
                                                    float* __restrict__ nrm) {
  int idx = blockIdx.x * 256 + threadIdx.x;   // 0..16383
  int sel = idx >> 13;
  int r   = idx & (NMAT - 1);
  const float* row = qkv + ((size_t)sel * NMAT + r) * 1024;
  float acc = 0.f;
  for (int f = 0; f < NF; ++f) { float t = row[f]; acc += t * t; }
  nrm[idx] = acc;
}

// ---------------------------------------------------------------------------
// Kernel 5: scores[b][j][i] = 1/(1+log1p(sqrt(max(q2[i]+k2[j]-2*kf.qf, 1e-12))))
// One wave per 16(j) x 64(i) strip: the kf A-operand is loaded once per K-step
// and reused by 4 WMMAs (4 accumulators). K=1024.
// ---------------------------------------------------------------------------
__global__ __launch_bounds__(32) void scores_kernel(const float* __restrict__ qkv,
                                                    const float* __restrict__ nrm,
                                                    float* __restrict__ scores) {
  const int lane = threadIdx.x;
  const int nIdx = lane & 15;
  const int hi   = lane >> 4;
  const int it = blockIdx.x, jt = blockIdx.y, b = blockIdx.z;
  const int i0 = it * 64, j0 = jt * 16;
  const float* qf = qkv + ((size_t)0 * NMAT + (size_t)b * NN) * 1024;
  const float* kf = qkv + ((size_t)1 * NMAT + (size_t)b * NN) * 1024;
  const float* arow  = kf + (size_t)(j0 + nIdx) * 1024;              // A: M = j
  const float* brow0 = qf + (size_t)(i0 +  0 + nIdx) * 1024;        // B: N = i
  const float* brow1 = qf + (size_t)(i0 + 16 + nIdx) * 1024;
  const float* brow2 = qf + (size_t)(i0 + 32 + nIdx) * 1024;
  const float* brow3 = qf + (size_t)(i0 + 48 + nIdx) * 1024;

  v8f acc0 = {}, acc1 = {}, acc2 = {}, acc3 = {};
  for (int k0 = hi * 2; k0 < NF; k0 += 4) {
    v2f a = *(const v2f*)(arow + k0);
    acc0 = WMMA_F32_16x16x4(a, *(const v2f*)(brow0 + k0), acc0);
    acc1 = WMMA_F32_16x16x4(a, *(const v2f*)(brow1 + k0), acc1);
    acc2 = WMMA_F32_16x16x4(a, *(const v2f*)(brow2 + k0), acc2);
    acc3 = WMMA_F32_16x16x4(a, *(const v2f*)(brow3 + k0), acc3);
  }

  float* sb = scores + (size_t)b * NN * NN;
#pragma unroll
  for (int t = 0; t < 4; ++t) {
    v8f acc = (t == 0) ? acc0 : ((t == 1) ? acc1 : ((t == 2) ? acc2 : acc3));
    int i = i0 + t * 16 + nIdx;
    float q2 = nrm[b * NN + i];
#pragma unroll
    for (int r = 0; r < 8; ++r) {
      int j = j0 + r + hi * 8;
      float d2 = q2 + nrm[NMAT + b * NN + j] - 2.f * acc[r];
      d2 = fmaxf(d2, 1e-12f);
      float e  = sqrtf(d2);
      float sc = 1.f / (1.f + log1pf(e));
      sb[(size_t)j * NN + i] = sc;
    }
  }
}

// ---------------------------------------------------------------------------
// Kernel 6: softmax over j (dim -2) of scores[b][j][i], in place.
// One thread per (b,i) column; coalesced across i.
// ---------------------------------------------------------------------------
__global__ __launch_bounds__(256) void softmax_kernel(float* __restrict__ scores) {
  int idx = blockIdx.x * 256 + threadIdx.x;   // 0..8191
  int b = idx >> 9, i = idx & (NN - 1);
  float* col = scores + (size_t)b * NN * NN + i;
  float mx = -1e30f;
  for (int j = 0; j < NN; ++j) mx = fmaxf(mx, col[(size_t)j * NN]);
  float sum = 0.f;
  for (int j = 0; j < NN; ++j) {
    float e = __expf(col[(size_t)j * NN] - mx);
    sum += e;
    col[(size_t)j * NN] = e;
  }
  float inv = 1.f / sum;
  for (int j = 0; j < NN; ++j) col[(size_t)j * NN] *= inv;
}

// ---------------------------------------------------------------------------
// Kernel 7: out[b][c][f] = sum_a p[b][c][a] * v[b][a][f]   (wmma, K=512)
// One wave per 64(c) x 16(f) strip: the strided v B-operand (2 scalar loads
// per K-step) is shared by 4 WMMAs; the contiguous p A-operand rows are cheap.
// ---------------------------------------------------------------------------
__global__ __launch_bounds__(32) void outgemm_kernel(const float* __restrict__ scores,
                                                     const float* __restrict__ qkv,
                                                     float* __restrict__ out) {
  const int lane = threadIdx.x;
  const int nIdx = lane & 15;
  const int hi   = lane >> 4;
  const int ft = blockIdx.x, ct = blockIdx.y, b = blockIdx.z;
  const int f0 = ft * 16, c0 = ct * 64;
  const float* p  = scores + (size_t)b * NN * NN;
  const float* vf = qkv + ((size_t)2 * NMAT + (size_t)b * NN) * 1024;
  const float* arow0 = p + (size_t)(c0 +  0 + nIdx) * NN;
  const float* arow1 = p + (size_t)(c0 + 16 + nIdx) * NN;
  const float* arow2 = p + (size_t)(c0 + 32 + nIdx) * NN;
  const float* arow3 = p + (size_t)(c0 + 48 + nIdx) * NN;

  v8f acc0 = {}, acc1 = {}, acc2 = {}, acc3 = {};
  for (int k0 = hi * 2; k0 < NN; k0 += 4) {
    v2f bb;
    bb.x = vf[(size_t)k0 * NF + f0 + nIdx];
    bb.y = vf[(size_t)(k0 + 1) * NF + f0 + nIdx];
    acc0 = WMMA_F32_16x16x4(*(const v2f*)(arow0 + k0), bb, acc0);
    acc1 = WMMA_F32_16x16x4(*(const v2f*)(arow1 + k0), bb, acc1);
    acc2 = WMMA_F32_16x16x4(*(const v2f*)(arow2 + k0), bb, acc2);
    acc3 = WMMA_F32_16x16x4(*(const v2f*)(arow3 + k0), bb, acc3);
  }

  float* ob = out + (size_t)b * NN * NF;
#pragma unroll
  for (int t = 0; t < 4; ++t) {
    v8f acc = (t == 0) ? acc0 : ((t == 1) ? acc1 : ((t == 2) ? acc2 : acc3));
#pragma unroll
    for (int r = 0; r < 8; ++r) {
      int c = c0 + t * 16 + r + hi * 8;
      ob[(size_t)c * NF + f0 + nIdx] = acc[r];
    }
  }
}

// ---------------------------------------------------------------------------
extern "C" void kernel_launch(void* const* d_in, const int* in_sizes, int n_in,
                              void* d_out, int out_size, void* d_ws, size_t ws_size,
                              hipStream_t stream) {
  (void)in_sizes; (void)n_in; (void)out_size; (void)ws_size;
  const float* x  = (const float*)d_in[0];   // [16,512,32,32]
  const float* qw = (const float*)d_in[1];   // [32,32]
  const float* kw = (const float*)d_in[2];
  const float* vw = (const float*)d_in[3];
  float* out = (float*)d_out;                // [16,512,32,32]

  float* ws     = (float*)d_ws;
  float* C      = ws;                                  // 3*1024
  float* xl     = C + 3 * 1024;                        // 8192*1024
  float* qkv    = xl + (size_t)NMAT * 1024;            // 3*8192*1024
  float* nrm    = qkv + (size_t)3 * NMAT * 1024;       // 16384
  float* scores = nrm + 2 * NMAT;                      // 16*512*512

  cayley_kernel<<<3, 256, 0, stream>>>(qw, kw, vw, C);
  jacobi_logm_kernel<<<NMAT, 32, 0, stream>>>(x, xl);
  transbase_kernel<<<dim3(NMAT, 3), 32, 0, stream>>>(xl, C, qkv);
  norms_kernel<<<64, 256, 0, stream>>>(qkv, nrm);
  scores_kernel<<<dim3(NN / 64, NN / 16, NB), 32, 0, stream>>>(qkv, nrm, scores);
  softmax_kernel<<<32, 256, 0, stream>>>(scores);
  outgemm_kernel<<<dim3(NF / 16, NN / 64, NB), 32, 0, stream>>>(scores, qkv, out);
}